// NTMMemory_5111011082619
// MI455X (gfx1250) — compile-verified
//
#include <hip/hip_runtime.h>
#include <math.h>

// ---------------------------------------------------------------------------
// NTM memory update, fused. B=64, N=2048, M=512, fp32.
// HBM floor: read memory (256MB) + write new_mem (256MB) => ~22us @ 23.3TB/s.
// Single streaming pass: new_mem elementwise on VALU (v_pk_fma_f32),
// dots = new_mem @ k on the matrix pipe (v_wmma_f32_16x16x4_f32, A = 16x4
// new_mem chunk in natural A-operand layout, B = k chunk replicated over
// columns), r = w^T @ new_mem accumulated per-lane in registers across 128
// rows and reduced cross-lane ONCE per block (amortizes the ds_bpermute
// butterfly 8x vs. per-tile reduction). NT cache hints keep the two 256MB
// streams out of the 192MB L2.
// ---------------------------------------------------------------------------

#define NTM_EPS 1e-8f
constexpr int Bn = 64;
constexpr int Nn = 2048;
constexpr int Mn = 512;
constexpr int ROWS_PER_BLOCK = 128;   // 8 sub-tiles of 16 rows

typedef __attribute__((ext_vector_type(2))) float v2f;
typedef __attribute__((ext_vector_type(8))) float v8f;

// ---- zero accumulators (harness poisons d_out / d_ws with 0xAA) -----------
__global__ void ntm_zero_kernel(float* __restrict__ dots,
                                float* __restrict__ sq,
                                float* __restrict__ r_out) {
  int i = blockIdx.x * blockDim.x + threadIdx.x;
  if (i < Bn * Nn) { dots[i] = 0.0f; sq[i] = 0.0f; }
  if (i < Bn * Mn) { r_out[i] = 0.0f; }
}

// ---- fused write + read + reduction pass ----------------------------------
// grid: Bn * (Nn/128) = 1024 blocks; block = 256 threads = 8 waves.
// Block owns (batch b, rows n0..n0+127). Wave w owns columns [w*64, w*64+64).
// Lane layout matches the WMMA A-operand: lane L<16 -> row (lane&15) of the
// sub-tile, K-cols {0,1} mod 4; lane L+16 -> same row, K-cols {2,3} mod 4.
__global__ __launch_bounds__(256) void ntm_update_kernel(
    const float* __restrict__ memory, const float* __restrict__ w_t,
    const float* __restrict__ e_t, const float* __restrict__ a_t,
    const float* __restrict__ k_t,
    float* __restrict__ r_out,       // (B,M) accumulated via atomics
    float* __restrict__ newmem_out,  // (B,N,M)
    float* __restrict__ dots,        // (B,N) ws, atomics
    float* __restrict__ sq) {        // (B,N) ws, atomics
  __shared__ float e_lds[Mn];
  __shared__ float a_lds[Mn];
  __shared__ float k_lds[Mn];

  const int b  = blockIdx.x >> 4;                 // 16 row-groups per batch
  const int n0 = (blockIdx.x & 15) * ROWS_PER_BLOCK;

  for (int i = threadIdx.x; i < Mn; i += 256) {
    e_lds[i] = e_t[b * Mn + i];
    a_lds[i] = a_t[b * Mn + i];
    k_lds[i] = k_t[b * Mn + i];
  }
  __syncthreads();

  const int lane = threadIdx.x & 31;
  const int wave = threadIdx.x >> 5;
  const int half = lane >> 4;                     // 0: K={0,1}, 1: K={2,3}
  const int colbase = wave * 64;

  // per-lane r partials for this wave's 64 columns, accumulated over all rows
  v2f racc[16] = {};

  for (int s = 0; s < 8; ++s) {                   // 8 sub-tiles of 16 rows
    const int row = n0 + s * 16 + (lane & 15);
    const float wn = w_t[b * Nn + row];
    const float* __restrict__ mrow = memory     + (size_t)(b * Nn + row) * Mn;
    float* __restrict__       orow = newmem_out + (size_t)(b * Nn + row) * Mn;

    v8f c = {};                                   // dots accumulator (per-row)
    float sqacc = 0.0f;

    #pragma unroll
    for (int ci = 0; ci < 16; ++ci) {
      const int m0 = colbase + 4 * ci;
      const int mc = m0 + 2 * half;               // this lane's 2 columns

      v2f mv = __builtin_nontemporal_load((const v2f*)(mrow + mc));
      const float ex = e_lds[mc], ey = e_lds[mc + 1];
      const float ax = a_lds[mc], ay = a_lds[mc + 1];

      // new_mem = mem*(1 - w*e) + w*a
      const float vx = fmaf(mv.x, 1.0f - wn * ex, wn * ax);
      const float vy = fmaf(mv.y, 1.0f - wn * ey, wn * ay);
      v2f ov; ov.x = vx; ov.y = vy;
      __builtin_nontemporal_store(ov, (v2f*)(orow + mc));

      // dots += new_mem(16x4) @ k(4x16-bcast) on the matrix pipe
      v2f A;  A.x = vx;          A.y = vy;
      v2f Bv; Bv.x = k_lds[mc];  Bv.y = k_lds[mc + 1];
      c = __builtin_amdgcn_wmma_f32_16x16x4_f32(
          /*neg_a=*/false, A, /*neg_b=*/false, Bv,
          /*c_mod=*/(short)0, c, /*reuse_a=*/false, /*reuse_b=*/false);

      // row squared-norm partial
      sqacc = fmaf(vx, vx, sqacc);
      sqacc = fmaf(vy, vy, sqacc);

      // r partials stay in registers; reduced once after all 128 rows
      racc[ci].x = fmaf(wn, vx, racc[ci].x);
      racc[ci].y = fmaf(wn, vy, racc[ci].y);
    }

    // squared norms: fold the two half-wave column sets, one atomic per row
    sqacc += __shfl_xor(sqacc, 16, 32);
    if (lane < 16) atomicAdd(&sq[b * Nn + row], sqacc);

    // dots: D VGPR r => sub-tile row r (lanes 0-15) / r+8 (lanes 16-31)
    const int nbase = n0 + s * 16;
    if (lane == 0) {
      #pragma unroll
      for (int rr = 0; rr < 8; ++rr)
        atomicAdd(&dots[b * Nn + nbase + rr], c[rr]);
    } else if (lane == 16) {
      #pragma unroll
      for (int rr = 0; rr < 8; ++rr)
        atomicAdd(&dots[b * Nn + nbase + 8 + rr], c[rr]);
    }
  }

  // r: lane index == row index, so a 16-lane xor-butterfly sums over rows.
  // Runs once per block (amortized over 128 rows).
  #pragma unroll
  for (int ci = 0; ci < 16; ++ci) {
    float rx = racc[ci].x, ry = racc[ci].y;
    #pragma unroll
    for (int msk = 1; msk <= 8; msk <<= 1) {
      rx += __shfl_xor(rx, msk, 32);
      ry += __shfl_xor(ry, msk, 32);
    }
    if ((lane & 15) == 0) {
      const int mc = colbase + 4 * ci + 2 * half;
      atomicAdd(&r_out[b * Mn + mc],     rx);
      atomicAdd(&r_out[b * Mn + mc + 1], ry);
    }
  }
}

// ---- content addressing: softmax_N(beta * cos) ----------------------------
// one block per batch; tiny (64*2048 elements total)
__global__ __launch_bounds__(256) void ntm_addr_kernel(
    const float* __restrict__ dots, const float* __restrict__ sq,
    const float* __restrict__ k_t, const float* __restrict__ beta_t,
    float* __restrict__ wc_out) {
  const int b = blockIdx.x;
  const int tid = threadIdx.x;
  __shared__ float red[256];
  __shared__ float xs[Nn];
  __shared__ float s_bcast;

  // ||k||
  float kp = 0.0f;
  for (int i = tid; i < Mn; i += 256) {
    float kv = k_t[b * Mn + i];
    kp = fmaf(kv, kv, kp);
  }
  red[tid] = kp; __syncthreads();
  for (int s = 128; s > 0; s >>= 1) {
    if (tid < s) red[tid] += red[tid + s];
    __syncthreads();
  }
  if (tid == 0) s_bcast = fmaxf(sqrtf(red[0]), NTM_EPS);
  __syncthreads();
  const float kden = s_bcast;
  const float beta = beta_t[b];

  // x = beta * cos ; track max
  float lmax = -1e30f;
  for (int nn = tid; nn < Nn; nn += 256) {
    const float mn = fmaxf(sqrtf(sq[b * Nn + nn]), NTM_EPS);
    const float x = beta * (dots[b * Nn + nn] / (mn * kden));
    xs[nn] = x;
    lmax = fmaxf(lmax, x);
  }
  red[tid] = lmax; __syncthreads();
  for (int s = 128; s > 0; s >>= 1) {
    if (tid < s) red[tid] = fmaxf(red[tid], red[tid + s]);
    __syncthreads();
  }
  if (tid == 0) s_bcast = red[0];
  __syncthreads();
  const float gmax = s_bcast;

  float lsum = 0.0f;
  for (int nn = tid; nn < Nn; nn += 256) {
    const float e = expf(xs[nn] - gmax);
    xs[nn] = e;
    lsum += e;
  }
  red[tid] = lsum; __syncthreads();
  for (int s = 128; s > 0; s >>= 1) {
    if (tid < s) red[tid] += red[tid + s];
    __syncthreads();
  }
  if (tid == 0) s_bcast = red[0];
  __syncthreads();
  const float inv = 1.0f / s_bcast;
  for (int nn = tid; nn < Nn; nn += 256)
    wc_out[b * Nn + nn] = xs[nn] * inv;
}

// ---------------------------------------------------------------------------
extern "C" void kernel_launch(void* const* d_in, const int* in_sizes, int n_in,
                              void* d_out, int out_size, void* d_ws, size_t ws_size,
                              hipStream_t stream) {
  const float* memory = (const float*)d_in[0];
  const float* w_t    = (const float*)d_in[1];
  const float* e_t    = (const float*)d_in[2];
  const float* a_t    = (const float*)d_in[3];
  const float* k_t    = (const float*)d_in[4];
  const float* beta_t = (const float*)d_in[5];

  float* out        = (float*)d_out;
  float* r_out      = out;                                   // B*M
  float* newmem_out = out + (size_t)Bn * Mn;                 // B*N*M
  float* wc_out     = out + (size_t)Bn * Mn + (size_t)Bn * Nn * Mn;  // B*N

  float* dots = (float*)d_ws;            // B*N
  float* sq   = dots + (size_t)Bn * Nn;  // B*N

  ntm_zero_kernel<<<(Bn * Nn + 255) / 256, 256, 0, stream>>>(dots, sq, r_out);
  ntm_update_kernel<<<Bn * (Nn / ROWS_PER_BLOCK), 256, 0, stream>>>(
      memory, w_t, e_t, a_t, k_t, r_out, newmem_out, dots, sq);
  ntm_addr_kernel<<<Bn, 256, 0, stream>>>(dots, sq, k_t, beta_t, wc_out);
}